// NonLinearModel_14525579395699
// MI455X (gfx1250) — compile-verified
//
#include <hip/hip_runtime.h>
#include <stdint.h>

// Problem dims from the reference.
#define BB   1024
#define TT   1024
#define DD   64
#define TPB  256
#define DIST 4      // async prefetch pipeline depth (power of 2)

// Memory-bound sequential scan: one thread owns one (b,d) chain.
// u_ext / noise are streamed through a DIST-deep async global->LDS pipeline
// (CDNA5 ASYNCcnt path); y is streamed out with coalesced stores.
__global__ __launch_bounds__(TPB) void nonlinear_scan_kernel(
    const float* __restrict__ x0,      // [B,1,D]
    const float* __restrict__ u_ext,   // [B,T,D]
    const float* __restrict__ noise,   // [B,T,D]
    const float* __restrict__ std_p,   // [1]
    float* __restrict__ y)             // [B,T,D]
{
    __shared__ float lds_u[DIST][TPB];
    __shared__ float lds_n[DIST][TPB];

    const int tid = blockIdx.x * TPB + threadIdx.x;   // 0 .. B*D-1
    const int b   = tid >> 6;                         // tid / D
    const int d   = tid & 63;                         // tid % D
    const float s = *std_p;

    // Element index of (b, t=0, d); per-t stride is D elements (256 B).
    const size_t base = ((size_t)b * TT) * DD + (size_t)d;
    uint64_t gu = (uint64_t)(uintptr_t)(u_ext + base);
    uint64_t gn = (uint64_t)(uintptr_t)(noise + base);
    float*   py = y + base;

    float x = x0[b * DD + d];

    // Wave-relative LDS byte addresses for this lane's slots.
    // (generic shared pointer truncated to 32 bits == LDS offset per the
    //  flat-aperture rule: the low 32 bits are the LDS address)
    uint32_t lu_addr[DIST], ln_addr[DIST];
    #pragma unroll
    for (int ss = 0; ss < DIST; ++ss) {
        lu_addr[ss] = (uint32_t)(uintptr_t)&lds_u[ss][threadIdx.x];
        ln_addr[ss] = (uint32_t)(uintptr_t)&lds_n[ss][threadIdx.x];
    }

    // ---- prologue: issue async prefetch for stages t = 0..DIST-1 ----
    #pragma unroll
    for (int ss = 0; ss < DIST; ++ss) {
        asm volatile("global_load_async_to_lds_b32 %0, %1, off"
                     :: "v"(lu_addr[ss]), "v"(gu + (uint64_t)(ss * DD * 4)) : "memory");
        asm volatile("global_load_async_to_lds_b32 %0, %1, off"
                     :: "v"(ln_addr[ss]), "v"(gn + (uint64_t)(ss * DD * 4)) : "memory");
    }
    gu += (uint64_t)(DIST * DD * 4);
    gn += (uint64_t)(DIST * DD * 4);

    // ---- steady state ----
    // Invariant at iter t: stages 0..t+DIST-1 issued (K = 2*(t+DIST) ops).
    // asynccnt <= 2*(DIST-1) ==> oldest 2*(t+1) ops complete (in-order) ==>
    // stage t has landed in LDS.
    for (int t = 0; t < TT - DIST; ++t) {
        const int buf = t & (DIST - 1);

        asm volatile("s_wait_asynccnt 6" ::: "memory");
        const float uu = lds_u[buf][threadIdx.x];
        const float nn = lds_n[buf][threadIdx.x];
        // Ensure the ds reads retired before the async engine overwrites buf.
        asm volatile("s_wait_dscnt 0" ::: "memory");

        // prefetch stage t+DIST into the buffer we just consumed
        asm volatile("global_load_async_to_lds_b32 %0, %1, off"
                     :: "v"(lu_addr[buf]), "v"(gu) : "memory");
        asm volatile("global_load_async_to_lds_b32 %0, %1, off"
                     :: "v"(ln_addr[buf]), "v"(gn) : "memory");
        gu += (uint64_t)(DD * 4);
        gn += (uint64_t)(DD * 4);

        *py = fmaf(nn, s, x);          // y[t] = x[t] + noise*std
        py += DD;
        x = x * x + 1.0f + uu;         // x[t+1]
    }

    // ---- epilogue: drain the last DIST stages ----
    asm volatile("s_wait_asynccnt 0" ::: "memory");
    #pragma unroll
    for (int t = TT - DIST; t < TT; ++t) {
        const int buf = t & (DIST - 1);
        const float uu = lds_u[buf][threadIdx.x];
        const float nn = lds_n[buf][threadIdx.x];
        *py = fmaf(nn, s, x);
        py += DD;
        x = x * x + 1.0f + uu;
    }
}

extern "C" void kernel_launch(void* const* d_in, const int* in_sizes, int n_in,
                              void* d_out, int out_size, void* d_ws, size_t ws_size,
                              hipStream_t stream) {
    (void)in_sizes; (void)n_in; (void)out_size; (void)d_ws; (void)ws_size;
    const float* x0    = (const float*)d_in[0];
    const float* u_ext = (const float*)d_in[1];
    const float* noise = (const float*)d_in[2];
    const float* sd    = (const float*)d_in[3];
    float* y = (float*)d_out;

    const int total  = BB * DD;          // 65536 independent chains
    const int blocks = total / TPB;      // 256 blocks x 256 threads (8 wave32/block)
    nonlinear_scan_kernel<<<blocks, TPB, 0, stream>>>(x0, u_ext, noise, sd, y);
}